// CostVolume_v2_41180146434758
// MI455X (gfx1250) — compile-verified
//
#include <hip/hip_runtime.h>
#include <stdint.h>

// ---------------------------------------------------------------------------
// CostVolume for stereo matching (pure data movement, store-BW bound).
//   out[b, c,     d, h, w] = (w >= d) ? x[b,c,h,w]     : 0
//   out[b, C + c, d, h, w] = (w >= d) ? y[b,c,h,w - d] : 0
// Shapes fixed by the reference: B=2, C=32, H=64, W=128; D = maxdisp+1 (49).
//
// Roofline: 0 FLOPs, ~205.5 MB stores -> ~8.8 us at 23.3 TB/s. Inputs (4 MB)
// are staged once per row into LDS via the CDNA5 async global->LDS path
// (ASYNCcnt) and reused D=49 times. Stores are wave-contiguous 512B bursts
// of global_store_b128.
// ---------------------------------------------------------------------------

#define B_ 2
#define C_ 32
#define H_ 64
#define W_ 128
#define BLOCK_ 256

typedef __attribute__((address_space(1))) void glob_void_t;
typedef __attribute__((address_space(3))) void lds_void_t;
typedef __attribute__((address_space(1))) int  glob_int_t;
typedef __attribute__((address_space(3))) int  lds_int_t;

// Async copy of one f32 from global memory into LDS (CDNA5 ASYNCcnt path).
__device__ __forceinline__ void async_load_f32_to_lds(const float* gp, float* lds_generic) {
#if __has_builtin(__builtin_amdgcn_global_load_async_to_lds_b32)
  // Toolchain signature (from diagnostics): (AS1 int*, AS3 int*, imm offset, imm cpol)
  __builtin_amdgcn_global_load_async_to_lds_b32(
      (glob_int_t*)(glob_void_t*)const_cast<float*>(gp),
      (lds_int_t*)(lds_void_t*)lds_generic,
      /*offset=*/0, /*cpol=*/0);
#else
  uint32_t lds_off = (uint32_t)(uintptr_t)lds_generic;
  asm volatile("global_load_async_to_lds_b32 %0, %1, off"
               :: "v"(lds_off), "v"(gp)
               : "memory");
#endif
}

__device__ __forceinline__ void wait_async_and_barrier() {
#if __has_builtin(__builtin_amdgcn_s_wait_asynccnt)
  __builtin_amdgcn_s_wait_asynccnt(0);
#else
  asm volatile("s_wait_asynccnt 0" ::: "memory");
#endif
  __syncthreads();
}

__global__ __launch_bounds__(BLOCK_) void cost_volume_kernel(
    const float* __restrict__ x, const float* __restrict__ y,
    const int* __restrict__ maxdisp_p, float* __restrict__ out) {
  __shared__ __align__(16) float srow[2 * W_];  // [0,128): x row, [128,256): y row

  const int tid = threadIdx.x;
  const int bch = blockIdx.x;                 // (b*C + c)*H + h
  const int h = bch & (H_ - 1);
  const int c = (bch >> 6) & (C_ - 1);
  const int b = bch >> 11;

  // ---- Stage one x-row and one y-row into LDS via async global->LDS ------
  const size_t rowbase = ((size_t)(b * C_ + c) * H_ + h) * (size_t)W_;
  {
    const int row = tid >> 7;                 // 0 -> x, 1 -> y
    const int col = tid & (W_ - 1);
    const float* gp = (row ? y : x) + rowbase + col;
    async_load_f32_to_lds(gp, &srow[tid]);    // 256 lanes -> 256 floats
  }
  wait_async_and_barrier();

  const int D = maxdisp_p[0] + 1;             // 49 for the reference

  const float* sx = srow;
  const float* sy = srow + W_;

  const size_t HW = (size_t)H_ * (size_t)W_;
  const size_t baseL = (size_t)(b * 2 * C_ + c) * (size_t)D * HW + (size_t)h * W_;
  const size_t baseR = baseL + (size_t)C_ * (size_t)D * HW;

  // ---- Per-thread assignment: lane = w4 (fixed), wave = d0; iterate d ----
  // Each wave (32 lanes) covers one full W=128 row per side per iteration:
  // contiguous 512B global_store_b128 bursts.
  const int w4 = tid & 31;                    // lane within wave
  const int d0 = tid >> 5;                    // wave id: 0..7
  const int w0 = w4 << 2;

  const float4 xa = *(const float4*)&sx[w0];  // loop-invariant ds_load_b128
  const float xv[4] = {xa.x, xa.y, xa.z, xa.w};

  float* pL = out + baseL + (size_t)d0 * HW + (size_t)w0;
  float* pR = out + baseR + (size_t)d0 * HW + (size_t)w0;
  const size_t step = (size_t)8 * HW;         // 8 waves stride the d axis

  for (int d = d0; d < D; d += 8) {
    float lv[4], rv[4];
#pragma unroll
    for (int j = 0; j < 4; ++j) {
      const int w = w0 + j;
      int yi = w - d;
      if (yi < 0) yi = 0;                     // keep LDS index in range
      const bool ok = (w >= d);
      lv[j] = ok ? xv[j] : 0.0f;
      rv[j] = ok ? sy[yi] : 0.0f;
    }
    *(float4*)pL = make_float4(lv[0], lv[1], lv[2], lv[3]);
    *(float4*)pR = make_float4(rv[0], rv[1], rv[2], rv[3]);
    pL += step;
    pR += step;
  }
}

extern "C" void kernel_launch(void* const* d_in, const int* in_sizes, int n_in,
                              void* d_out, int out_size, void* d_ws, size_t ws_size,
                              hipStream_t stream) {
  (void)in_sizes; (void)n_in; (void)out_size; (void)d_ws; (void)ws_size;

  const float* x  = (const float*)d_in[0];
  const float* y  = (const float*)d_in[1];
  const int*   md = (const int*)d_in[2];
  float*       out = (float*)d_out;

  dim3 grid(B_ * C_ * H_);   // 4096 blocks, one (b,c,h) row pair each
  dim3 block(BLOCK_);        // 8 wave32 waves
  cost_volume_kernel<<<grid, block, 0, stream>>>(x, y, md, out);
}